// MSEHingeLoss_65532611002861
// MI455X (gfx1250) — compile-verified
//
#include <hip/hip_runtime.h>
#include <hip/hip_bf16.h>

typedef __attribute__((ext_vector_type(2))) float v2f;
typedef __attribute__((ext_vector_type(8))) float v8f;

#define BLK 256   // threads per block (8 wave32)
#define TI  4     // i-values per thread
#define JT  256   // j-slice per block staged in LDS

// Exact wave32 sum using V_WMMA_F32_16X16X4_F32.
// A layout (16x4 f32): lanes 0-15 -> M=lane, VGPR0=K0, VGPR1=K1;
//                      lanes 16-31 -> M=lane-16, VGPR0=K2, VGPR1=K3.
// Put s in VGPR0, 0 in VGPR1, B = ones  =>  D[m][n] = s_m + s_{m+16}.
// C/D layout: lane n holds rows {0..7} (n<16) or {8..15} (n>=16) in 8 VGPRs,
// so per-lane sum of the 8 VGPRs + shfl_xor(16) gives the full 32-lane total
// in every lane. Requires EXEC == all ones (callers guarantee no divergence).
__device__ __forceinline__ float wave_reduce_wmma(float s) {
    v2f a; a[0] = s;    a[1] = 0.0f;
    v2f b; b[0] = 1.0f; b[1] = 1.0f;
    v8f c = {};
    c = __builtin_amdgcn_wmma_f32_16x16x4_f32(
        /*neg_a=*/false, a, /*neg_b=*/false, b,
        /*c_mod=*/(short)0, c, /*reuse_a=*/false, /*reuse_b=*/false);
    float r = ((c[0] + c[1]) + (c[2] + c[3])) + ((c[4] + c[5]) + (c[6] + c[7]));
    r += __shfl_xor(r, 16, 32);
    return r;
}

__global__ void __launch_bounds__(BLK)
pair_loss_kernel(const float* __restrict__ inp, const float* __restrict__ tgt,
                 float* __restrict__ partials, int n, int njb) {
    __shared__ float2 tile[JT];           // {b_j, t_j}
    __shared__ float  wpart[BLK / 32];

    const int it  = (int)blockIdx.x / njb;
    const int jt  = (int)blockIdx.x % njb;
    const int tid = (int)threadIdx.x;

    // Stage this block's j-slice into LDS: b_j = 0.5*in[j], t_j = tgt[j].
    // OOB j gets t = -inf so the mask (t_j > t_i) is always false.
    {
        int j = jt * JT + tid;
        float2 e;
        if (j < n) { float x = inp[j]; e.x = 0.5f * x; e.y = tgt[j]; }
        else       { e.x = 0.0f; e.y = -__builtin_inff(); }
        tile[tid] = e;
    }

    // Per-thread i values: a_i = 0.5*in_i + 0.5*(in_i - tg_i)^2 + 0.005.
    // OOB i gets t_i = +inf so the mask is always false (no divergence later).
    float a[TI], ti[TI];
#pragma unroll
    for (int k = 0; k < TI; ++k) {
        int i = it * (BLK * TI) + k * BLK + tid;
        if (i < n) {
            float x = inp[i], t = tgt[i];
            float d = x - t;
            a[k]  = 0.5f * x + 0.5f * d * d + 0.005f;
            ti[k] = t;
        } else {
            a[k]  = 0.0f;
            ti[k] = __builtin_inff();
        }
    }
    __syncthreads();

    float s[TI];
#pragma unroll
    for (int k = 0; k < TI; ++k) s[k] = 0.0f;

#pragma unroll 4
    for (int jj = 0; jj < JT; ++jj) {
        float2 bt = tile[jj];             // LDS broadcast (same addr all lanes)
#pragma unroll
        for (int k = 0; k < TI; ++k) {
            float d = a[k] - bt.x;        // a_i - b_j
            d = fmaxf(d, 0.0f);           // relu
            s[k] += (bt.y > ti[k]) ? d : 0.0f;   // pair mask
        }
    }

    float stot = (s[0] + s[1]) + (s[2] + s[3]);
    float wsum = wave_reduce_wmma(stot);  // all 256 threads reach here (EXEC full)

    if ((tid & 31) == 0) wpart[tid >> 5] = wsum;
    __syncthreads();
    if (tid == 0) {
        float r = 0.0f;
#pragma unroll
        for (int w = 0; w < BLK / 32; ++w) r += wpart[w];
        partials[blockIdx.x] = r;
    }
}

__global__ void __launch_bounds__(BLK)
finalize_kernel(const float* __restrict__ partials, int P,
                float* __restrict__ out, float scale) {
    __shared__ float wpart[BLK / 32];
    const int tid = (int)threadIdx.x;

    float s = 0.0f;
    for (int idx = tid; idx < P; idx += BLK) s += partials[idx];  // fixed order

    float wsum = wave_reduce_wmma(s);     // all threads participate

    if ((tid & 31) == 0) wpart[tid >> 5] = wsum;
    __syncthreads();
    if (tid == 0) {
        float r = 0.0f;
#pragma unroll
        for (int w = 0; w < BLK / 32; ++w) r += wpart[w];
        out[0] = scale * r;
    }
}

extern "C" void kernel_launch(void* const* d_in, const int* in_sizes, int n_in,
                              void* d_out, int out_size, void* d_ws, size_t ws_size,
                              hipStream_t stream) {
    const float* inp = (const float*)d_in[0];
    const float* tgt = (const float*)d_in[1];
    float* out       = (float*)d_out;
    float* partials  = (float*)d_ws;

    const int n   = in_sizes[0];
    const int nib = (n + BLK * TI - 1) / (BLK * TI);   // i-tiles (8 for N=8192)
    const int njb = (n + JT - 1) / JT;                 // j-tiles (32 for N=8192)
    const int P   = nib * njb;                         // 256 block partials

    pair_loss_kernel<<<dim3(P), dim3(BLK), 0, stream>>>(inp, tgt, partials, n, njb);

    const float scale = 2.0f / ((float)n * (float)n);
    finalize_kernel<<<dim3(1), dim3(BLK), 0, stream>>>(partials, P, out, scale);
}